// SupernodeLearn_29068338659474
// MI455X (gfx1250) — compile-verified
//
#include <hip/hip_runtime.h>

#define DIM 128
#define PPITCH 288  // LDS pair-row pitch in floats: 288 % 64 == 32 -> for a b64
                    // B-fragment load, lanes 0-15 and 16-31 use disjoint bank halves

typedef __attribute__((ext_vector_type(2))) float v2f;
typedef __attribute__((ext_vector_type(8))) float v8f;

// ---------------------------------------------------------------- utilities
__global__ __launch_bounds__(256) void k_zero4(float4* __restrict__ p, int n4) {
  int i = blockIdx.x * 256 + threadIdx.x;
  if (i < n4) p[i] = make_float4(0.f, 0.f, 0.f, 0.f);
}

__global__ __launch_bounds__(256) void k_deg(const int* __restrict__ src,
                                             float* __restrict__ deg, int ne) {
  int i = blockIdx.x * 256 + threadIdx.x;
  if (i < ne) atomicAdd(&deg[src[i]], 1.0f);
}

__global__ __launch_bounds__(256) void k_norm(float* __restrict__ norm, int n) {
  int i = blockIdx.x * 256 + threadIdx.x;
  if (i < n) {
    float d = norm[i];
    norm[i] = rsqrtf(d < 1.0f ? 1.0f : d);
  }
}

// one wave32 per edge: coalesced 512B gather, src-side norm^P fused, 4 f32 atomics/lane
template <int P>
__global__ __launch_bounds__(256) void k_scatter(const float* __restrict__ x,
                                                 const int* __restrict__ src,
                                                 const int* __restrict__ dst,
                                                 const float* __restrict__ normv,
                                                 float* __restrict__ y, int ne) {
  int gid = blockIdx.x * 256 + threadIdx.x;
  int e = gid >> 5;
  if (e >= ne) return;
  int lane = gid & 31;
  int s = src[e];
  int d = dst[e];
  float sc = normv[s];
  if (P == 2) sc *= sc;
  float4 v = ((const float4*)(x + (size_t)s * DIM))[lane];
  v.x *= sc; v.y *= sc; v.z *= sc; v.w *= sc;
  float* o = y + (size_t)d * DIM + lane * 4;
  atomicAdd(o + 0, v.x);
  atomicAdd(o + 1, v.y);
  atomicAdd(o + 2, v.z);
  atomicAdd(o + 3, v.w);
}

// ----------------------------------------------------------------- GEMM
// out[tile] = (SCALEA ? norm∘X : X) @ W + bias (+ acc) (+ relu), via
// V_WMMA_F32_16X16X4_F32; dst-side norm applied to accumulators (row scaling
// commutes with the GEMM). W staged in LDS pre-swizzled so each B fragment is
// a single ds_load_b64: pair-row p=k/2 holds [W[2p][c], W[2p+1][c]] adjacent.
template <int ACC, int RELU, int SCALEA>
__global__ __launch_bounds__(256) void k_gemm(
    const float* __restrict__ X, const float* __restrict__ W,
    const float* __restrict__ bias, const float* __restrict__ normv,
    const float* __restrict__ acc, float* __restrict__ out, int nTiles)
{
  __shared__ float lW[64 * PPITCH];
  __shared__ float lB[DIM];

  int t = threadIdx.x;
  for (int i4 = t; i4 < DIM * DIM / 4; i4 += 256) {
    int lin = i4 * 4;
    int k = lin >> 7;
    int col = lin & 127;
    float4 w = ((const float4*)W)[i4];
    float* p = &lW[(k >> 1) * PPITCH + 2 * col + (k & 1)];
    p[0] = w.x; p[2] = w.y; p[4] = w.z; p[6] = w.w;
  }
  if (t < DIM) lB[t] = bias[t];
  __syncthreads();

  int wave = t >> 5;
  int lane = t & 31;
  int rowTile = blockIdx.x * 8 + wave;
  if (rowTile >= nTiles) return;  // whole wave exits together -> EXEC all-ones for WMMA

  int m  = lane & 15;
  int hi = lane >> 4;
  const float* xrow = X + ((size_t)rowTile * 16 + m) * DIM + 2 * hi;

  v8f zero = {};
  v8f c[8];
#pragma unroll
  for (int nt = 0; nt < 8; ++nt) c[nt] = zero;

#pragma unroll 2
  for (int k = 0; k < DIM; k += 4) {
    // A: lane(hi,m) holds X[m][k+2hi], X[m][k+2hi+1] -- one global b64 load
    float2 av = *(const float2*)(xrow + k);
    v2f a;
    a.x = av.x;
    a.y = av.y;
    // B: pair-row (k/2 + hi) gives this lane K=k+2hi,k+2hi+1 -- one ds_load_b64
    const float2* wp = (const float2*)&lW[((k >> 1) + hi) * PPITCH];
#pragma unroll
    for (int nt = 0; nt < 8; ++nt) {
      float2 bv2 = wp[nt * 16 + m];
      v2f b;
      b.x = bv2.x;
      b.y = bv2.y;
      c[nt] = __builtin_amdgcn_wmma_f32_16x16x4_f32(
          false, a, false, b, (short)0, c[nt], false, false);
    }
  }

  // C/D layout: VGPR r -> row 8*hi + r, col nt*16 + m
  size_t rowBase = (size_t)rowTile * 16 + 8 * hi;
  float nr[8];
#pragma unroll
  for (int r = 0; r < 8; ++r) nr[r] = SCALEA ? normv[rowBase + r] : 1.0f;

#pragma unroll
  for (int nt = 0; nt < 8; ++nt) {
    int col = nt * 16 + m;
    float bv = lB[col];
    float* po = out + rowBase * DIM + col;
    const float* pa = acc + rowBase * DIM + col;
#pragma unroll
    for (int r = 0; r < 8; ++r) {
      float v = SCALEA ? fmaf(c[nt][r], nr[r], bv) : (c[nt][r] + bv);
      if (ACC) v += pa[r * DIM];
      if (RELU) v = v > 0.f ? v : 0.f;
      po[r * DIM] = v;
    }
  }
}

// ------------------------------------------------------------- host driver
static void run_layer(const float* X, const float* W, const float* b,
                      const int* src, const int* dst, const float* norm,
                      float* aggA, float* aggB, float* rst, float* OUT,
                      int N, int E, hipStream_t stream)
{
  int n4 = N * DIM / 4;
  int nTiles = N / 16;
  int gemmGrid = (nTiles + 7) / 8;
  int g4 = (n4 + 255) / 256;
  int gE = (E * 32 + 255) / 256;

  // rst = X @ W + b
  k_gemm<0, 0, 0><<<gemmGrid, 256, 0, stream>>>(X, W, b, norm, rst, rst, nTiles);

  // hop 1: aggA = scatter(dst, norm[src] * X[src]); rst += (norm∘aggA) @ W + b
  k_zero4<<<g4, 256, 0, stream>>>((float4*)aggA, n4);
  k_scatter<1><<<gE, 256, 0, stream>>>(X, src, dst, norm, aggA, E);
  k_gemm<1, 0, 1><<<gemmGrid, 256, 0, stream>>>(aggA, W, b, norm, rst, rst, nTiles);

  // hop 2: h1 = norm∘aggA, so scatter source = norm[src]^2 * aggA[src]
  k_zero4<<<g4, 256, 0, stream>>>((float4*)aggB, n4);
  k_scatter<2><<<gE, 256, 0, stream>>>(aggA, src, dst, norm, aggB, E);
  // OUT = relu(rst + (norm∘aggB) @ W + b)
  k_gemm<1, 1, 1><<<gemmGrid, 256, 0, stream>>>(aggB, W, b, norm, rst, OUT, nTiles);
}

extern "C" void kernel_launch(void* const* d_in, const int* in_sizes, int n_in,
                              void* d_out, int out_size, void* d_ws, size_t ws_size,
                              hipStream_t stream)
{
  const float* feat = (const float*)d_in[0];
  const float* W1   = (const float*)d_in[1];
  const float* b1   = (const float*)d_in[2];
  const float* W2   = (const float*)d_in[3];
  const float* b2   = (const float*)d_in[4];
  const int*   src  = (const int*)d_in[5];
  const int*   dst  = (const int*)d_in[6];
  float* out = (float*)d_out;

  int N = in_sizes[0] / DIM;   // 50000
  int E = in_sizes[5];         // 800000

  // workspace: norm (N f32, 256B-aligned pad) + 3 dense N x D buffers (~77 MB)
  char* ws = (char*)d_ws;
  float* norm = (float*)ws;
  size_t off = (((size_t)N * 4) + 255) & ~(size_t)255;
  float* aggA = (float*)(ws + off);
  float* aggB = aggA + (size_t)N * DIM;
  float* rst  = aggB + (size_t)N * DIM;

  // degree (over src) -> norm = rsqrt(max(deg,1)); shared by both layers
  int nd4 = (N + 3) / 4;
  k_zero4<<<(nd4 + 255) / 256, 256, 0, stream>>>((float4*)norm, nd4);
  k_deg<<<(E + 255) / 256, 256, 0, stream>>>(src, norm, E);
  k_norm<<<(N + 255) / 256, 256, 0, stream>>>(norm, N);

  // layer 1 writes h1 into d_out; layer 2 consumes it and overwrites d_out last
  run_layer(feat, W1, b1, src, dst, norm, aggA, aggB, rst, out, N, E, stream);
  run_layer(out,  W2, b2, src, dst, norm, aggA, aggB, rst, out, N, E, stream);
}